// DynamicPartitionMaskStitchModule_63599875719267
// MI455X (gfx1250) — compile-verified
//
#include <hip/hip_runtime.h>
#include <stdint.h>

// DynamicPartition + MaskStitch composed == identity permutation:
//   out[order[i]] = data[order[i]]  with `order` a bijection  =>  out = data.
// Optimal MI455X kernel is a pure streaming copy of 512 MB (roofline ~44us at
// 23.3 TB/s). We use the CDNA5 async global<->LDS data mover path
// (GLOBAL_LOAD_ASYNC_TO_LDS_B128 / GLOBAL_STORE_ASYNC_FROM_LDS_B128, ASYNCcnt)
// for the bulk, with plain b128/scalar copies for generic remainders.

#define BLOCK   256   // 8 wave32 waves per block
#define K_SLOTS 8     // float4 slots per thread per iteration (32 KB LDS/block)

__global__ __launch_bounds__(BLOCK)
void DynamicPartitionMaskStitch_identity_copy(const float4* __restrict__ src,
                                              float4* __restrict__ dst,
                                              size_t n4, size_t iter_full)
{
    __shared__ float4 smem[BLOCK * K_SLOTS];   // 32 KB staging

    const uint32_t tid      = threadIdx.x;
    const size_t   blk_base = (size_t)blockIdx.x * (BLOCK * K_SLOTS);
    const size_t   chunk    = (size_t)gridDim.x * (BLOCK * K_SLOTS);

    // Workgroup-relative LDS byte offsets for this lane's slots.
    // Generic LDS aperture addresses keep the LDS byte offset in bits [31:0]
    // (ISA 10.2 aperture mapping), so truncation yields the DS address the
    // async instructions expect.
    uint32_t lds_addr[K_SLOTS];
#pragma unroll
    for (int k = 0; k < K_SLOTS; ++k)
        lds_addr[k] = (uint32_t)(uintptr_t)(&smem[k * BLOCK + tid]);

    const void* sbase = (const void*)src;
    void*       dbase = (void*)dst;

    for (size_t it = 0; it < iter_full; ++it) {
        const size_t base = it * chunk + blk_base;   // in float4 units

        // Phase 1: async global -> LDS, 16 B per lane per slot (ASYNCcnt).
#pragma unroll
        for (int k = 0; k < K_SLOTS; ++k) {
            const uint32_t goff =
                (uint32_t)((base + (size_t)(k * BLOCK) + tid) * sizeof(float4));
            asm volatile("global_load_async_to_lds_b128 %0, %1, %2"
                         :
                         : "v"(lds_addr[k]), "v"(goff), "s"(sbase)
                         : "memory");
        }
        // Load data must be resident in LDS before the async stores read it.
        asm volatile("s_wait_asynccnt 0x0" ::: "memory");

        // Phase 2: async LDS -> global.
#pragma unroll
        for (int k = 0; k < K_SLOTS; ++k) {
            const uint32_t goff =
                (uint32_t)((base + (size_t)(k * BLOCK) + tid) * sizeof(float4));
            asm volatile("global_store_async_from_lds_b128 %0, %1, %2"
                         :
                         : "v"(goff), "v"(lds_addr[k]), "s"(dbase)
                         : "memory");
        }
        // Stores must drain LDS before next iteration overwrites the slots.
        asm volatile("s_wait_asynccnt 0x0" ::: "memory");
    }

    // Generic remainder (empty for the 500000x256 shape): plain b128 copies.
    const size_t done    = iter_full * chunk;
    const size_t gstride = (size_t)gridDim.x * BLOCK;
    for (size_t i = done + (size_t)blockIdx.x * BLOCK + tid; i < n4; i += gstride)
        dst[i] = src[i];
}

// Scalar tail for element counts not divisible by 4 (empty here: 128M % 4 == 0).
__global__ void DynamicPartitionMaskStitch_tail(const float* __restrict__ src,
                                                float* __restrict__ dst,
                                                size_t n, size_t start)
{
    size_t i = start + (size_t)blockIdx.x * blockDim.x + threadIdx.x;
    if (i < n) dst[i] = src[i];
}

extern "C" void kernel_launch(void* const* d_in, const int* in_sizes, int n_in,
                              void* d_out, int out_size, void* d_ws, size_t ws_size,
                              hipStream_t stream)
{
    (void)n_in; (void)d_ws; (void)ws_size; (void)out_size;

    const float* data = (const float*)d_in[0];   // (500000, 256) fp32
    // d_in[1] (partitions) is provably unused: the op composes to identity.
    float* out = (float*)d_out;

    const size_t n  = (size_t)in_sizes[0];       // 128,000,000 floats
    const size_t n4 = n / 4;                     // 32,000,000 float4

    // 3125 blocks * 256 threads * 8 slots = 6,400,000 float4 per grid pass;
    // 32,000,000 / 6,400,000 = 5 exact full iterations for this shape.
    int grid = (int)((n4 + (size_t)(BLOCK * K_SLOTS) - 1) / (BLOCK * K_SLOTS));
    if (grid > 3125) grid = 3125;
    if (grid < 1)    grid = 1;

    const size_t chunk     = (size_t)grid * BLOCK * K_SLOTS;
    const size_t iter_full = n4 / chunk;

    DynamicPartitionMaskStitch_identity_copy<<<grid, BLOCK, 0, stream>>>(
        (const float4*)data, (float4*)out, n4, iter_full);

    const size_t vec_end = n4 * 4;
    if (vec_end < n) {
        const size_t rem = n - vec_end;
        DynamicPartitionMaskStitch_tail<<<(unsigned)((rem + 255) / 256), 256, 0, stream>>>(
            data, out, n, vec_end);
    }
}